// BaselineGNN_88218628260833
// MI455X (gfx1250) — compile-verified
//
#include <hip/hip_runtime.h>
#include <hip/hip_bf16.h>

// ---------------- problem constants (match reference) ----------------
#define N_NODES   100000
#define N_EDGES   1600000
#define N_GRAPHS  4096
#define DIM       128
#define HDIM      64
#define N_LAYERS  3
#define BN_EPS    1e-5f

typedef __attribute__((ext_vector_type(2))) float v2f;
typedef __attribute__((ext_vector_type(8))) float v8f;

// ---------------- small utility kernels ----------------
__global__ void k_zero(float* __restrict__ p, long long n) {
    long long i = (long long)blockIdx.x * blockDim.x + threadIdx.x;
    if (i < n) p[i] = 0.0f;
}

__global__ void k_copy4(const float4* __restrict__ s, float4* __restrict__ d, long long n4) {
    long long i = (long long)blockIdx.x * blockDim.x + threadIdx.x;
    if (i < n4) d[i] = s[i];
}

// ---------------- degree ----------------
__global__ void k_deg(const int* __restrict__ dst, float* __restrict__ deg) {
    int e = blockIdx.x * blockDim.x + threadIdx.x;
    if (e < N_EDGES) atomicAdd(&deg[dst[e]], 1.0f);
}

__global__ void k_invdeg(float* __restrict__ deg) {
    int i = blockIdx.x * blockDim.x + threadIdx.x;
    if (i < N_NODES) deg[i] = 1.0f / fmaxf(deg[i], 1.0f);
}

// ---------------- edge scatter: agg[dst] += x[src]  (L2-resident atomics) ----------------
__global__ void k_scatter(const int* __restrict__ src, const int* __restrict__ dst,
                          const float* __restrict__ x, float* __restrict__ agg) {
    long long idx = (long long)blockIdx.x * blockDim.x + threadIdx.x;
    if (idx >= (long long)N_EDGES * 32) return;
    int e  = (int)(idx >> 5);
    int c4 = (int)(idx & 31);
    int s = src[e];
    int d = dst[e];
    float4 v = ((const float4*)x)[(long long)s * 32 + c4];
    float* a = agg + (long long)d * DIM + c4 * 4;
    atomicAdd(a + 0, v.x);
    atomicAdd(a + 1, v.y);
    atomicAdd(a + 2, v.z);
    atomicAdd(a + 3, v.w);
}

// ---------------- weight pre-pack into WMMA B-fragment order ----------------
// Wcat = [Wl ; Wr] is [256 x 128].  For K-step s (K = 4s..4s+3), column tile wid,
// lane holds B rows {4s+2*half, 4s+2*half+1} at col wid*16+laneN  ->  one float2
// per lane, laid out so each wave reads 32 consecutive float2 (256 B, coalesced).
__global__ void k_packW(const float* __restrict__ Wl, const float* __restrict__ Wr,
                        float2* __restrict__ Wp) {
    int idx = blockIdx.x * blockDim.x + threadIdx.x;      // 64 * 8 * 32
    if (idx >= 64 * 8 * 32) return;
    int lane  = idx & 31;
    int wid   = (idx >> 5) & 7;
    int s     = idx >> 8;
    int half  = lane >> 4;
    int laneN = lane & 15;
    int kk  = 4 * s + 2 * half;                           // kk, kk+1 stay in same half
    int col = wid * 16 + laneN;
    const float* W = (kk < DIM) ? (Wl + (long long)kk * DIM)
                                : (Wr + (long long)(kk - DIM) * DIM);
    float2 v;
    v.x = W[col];
    v.y = W[DIM + col];
    Wp[idx] = v;
}

// ---------------- fused GEMM: y = (agg*invdeg) @ Wl + x @ Wr + b ----------------
// Block: 256 thr = 8 waves; 64 rows x 128 cols.  A = [agg | x] rows staged into LDS
// with global_load_async_to_lds_b128 (ASYNCcnt path); inv_deg folded into the
// A-fragment at use time (VALU co-executes with the WMMA pipe).
#define ROWS_PER_BLK 64
#define LDA 260   // 256 + 4 pad: A-frag ds_load_b64 lands on distinct banks

__global__ __launch_bounds__(256) void k_gemm_wmma(
    const float* __restrict__ agg, const float* __restrict__ xin,
    const float* __restrict__ invdeg, const float2* __restrict__ Wpack,
    const float* __restrict__ bias, float* __restrict__ y)
{
    __shared__ float As[ROWS_PER_BLK * LDA];

    const int tid = threadIdx.x;
    const int r0  = blockIdx.x * ROWS_PER_BLK;

    // ---- async stage of A tile: 64 rows x 256 floats = 4096 x 16B chunks ----
    // LDS offset for chunk (r, c4) is r*1040 + c4*16 for BOTH halves
    // (c4<32 -> agg cols, c4>=32 -> x cols); rows are 16B-aligned (1040 % 16 == 0).
    {
        const unsigned ldsBase = (unsigned)(size_t)(&As[0]);
        for (int it = tid; it < ROWS_PER_BLK * 64; it += 256) {
            int r  = it >> 6;
            int c4 = it & 63;
            long long grow = (long long)(r0 + r);          // OOB rows land in adjacent
            const float* gsrc = (c4 < 32)                  // ws buffers (allocated, rows
                ? agg + grow * DIM + c4 * 4                // never stored) -> safe
                : xin + grow * DIM + (c4 - 32) * 4;
            unsigned laddr = ldsBase + (unsigned)(r * (LDA * 4) + c4 * 16);
            asm volatile("global_load_async_to_lds_b128 %0, %1, off"
                         :: "v"(laddr), "v"(gsrc) : "memory");
        }
        asm volatile("s_wait_asynccnt 0x0" ::: "memory");
    }
    __syncthreads();

    const int wid   = tid >> 5;
    const int lane  = tid & 31;
    const int half  = lane >> 4;
    const int laneN = lane & 15;
    const int col   = wid * 16 + laneN;

    // per-lane inverse-degree for the 4 row-tiles this wave touches
    float idg[4];
    #pragma unroll
    for (int rt = 0; rt < 4; ++rt) {
        int gr = r0 + rt * 16 + laneN;
        idg[rt] = (gr < N_NODES) ? invdeg[gr] : 0.0f;
    }

    v8f acc[4];
    #pragma unroll
    for (int rt = 0; rt < 4; ++rt) acc[rt] = v8f{};

    const int kbase = 2 * half;

    // ---- K-steps 0..31: agg half, scale A-frag by inv_deg ----
    for (int s = 0; s < 32; ++s) {
        v2f bfrag = *(const v2f*)&Wpack[(s * 8 + wid) * 32 + lane];
        const int kk = 4 * s + kbase;
        #pragma unroll
        for (int rt = 0; rt < 4; ++rt) {
            v2f a = *(const v2f*)&As[(rt * 16 + laneN) * LDA + kk];
            a.x *= idg[rt];
            a.y *= idg[rt];
            acc[rt] = __builtin_amdgcn_wmma_f32_16x16x4_f32(
                false, a, false, bfrag, (short)0, acc[rt], false, false);
        }
    }
    // ---- K-steps 32..63: x half, unscaled ----
    for (int s = 32; s < 64; ++s) {
        v2f bfrag = *(const v2f*)&Wpack[(s * 8 + wid) * 32 + lane];
        const int kk = 4 * s + kbase;
        #pragma unroll
        for (int rt = 0; rt < 4; ++rt) {
            v2f a = *(const v2f*)&As[(rt * 16 + laneN) * LDA + kk];
            acc[rt] = __builtin_amdgcn_wmma_f32_16x16x4_f32(
                false, a, false, bfrag, (short)0, acc[rt], false, false);
        }
    }

    // ---- epilogue: add bias, store (C/D layout: VGPR j -> M=j / M=j+8) ----
    const float bcol = bias[col];
    #pragma unroll
    for (int rt = 0; rt < 4; ++rt) {
        #pragma unroll
        for (int j = 0; j < 8; ++j) {
            int row = r0 + rt * 16 + j + 8 * half;
            if (row < N_NODES)
                y[(long long)row * DIM + col] = acc[rt][j] + bcol;
        }
    }
}

// ---------------- BatchNorm: column sums / sumsq ----------------
__global__ void k_bnstats(const float* __restrict__ y, float* __restrict__ sums) {
    int c = threadIdx.x;              // 128 threads, one column each (coalesced)
    float s = 0.f, s2 = 0.f;
    for (int r = blockIdx.x; r < N_NODES; r += gridDim.x) {
        float v = y[(long long)r * DIM + c];
        s  += v;
        s2 += v * v;
    }
    atomicAdd(&sums[c],       s);
    atomicAdd(&sums[DIM + c], s2);
}

__global__ void k_bnfinal(const float* __restrict__ sums,
                          const float* __restrict__ gamma, const float* __restrict__ beta,
                          float* __restrict__ ss) {
    int c = threadIdx.x;
    if (c < DIM) {
        float inv_n = 1.0f / (float)N_NODES;
        float mu  = sums[c] * inv_n;
        float var = sums[DIM + c] * inv_n - mu * mu;
        float sc  = gamma[c] * rsqrtf(var + BN_EPS);
        ss[c]       = sc;                 // scale
        ss[DIM + c] = beta[c] - mu * sc;  // shift
    }
}

// x_next = relu(y * scale + shift), vectorized float4
__global__ void k_bnapply(const float* __restrict__ y, const float* __restrict__ ss,
                          float* __restrict__ xout) {
    long long i = (long long)blockIdx.x * blockDim.x + threadIdx.x;  // float4 index
    if (i >= (long long)N_NODES * 32) return;
    int c4 = (int)(i & 31);
    float4 v  = ((const float4*)y)[i];
    float4 sc = ((const float4*)ss)[c4];
    float4 sh = ((const float4*)(ss + DIM))[c4];
    float4 o;
    o.x = fmaxf(v.x * sc.x + sh.x, 0.f);
    o.y = fmaxf(v.y * sc.y + sh.y, 0.f);
    o.z = fmaxf(v.z * sc.z + sh.z, 0.f);
    o.w = fmaxf(v.w * sc.w + sh.w, 0.f);
    ((float4*)xout)[i] = o;
}

// ---------------- global mean pool ----------------
__global__ void k_cnt(const int* __restrict__ batch, float* __restrict__ cnt) {
    int i = blockIdx.x * blockDim.x + threadIdx.x;
    if (i < N_NODES) atomicAdd(&cnt[batch[i]], 1.0f);
}

__global__ void k_pool(const int* __restrict__ batch, const float* __restrict__ x,
                       float* __restrict__ pooled) {
    long long idx = (long long)blockIdx.x * blockDim.x + threadIdx.x;
    if (idx >= (long long)N_NODES * 32) return;
    int i  = (int)(idx >> 5);
    int c4 = (int)(idx & 31);
    int g  = batch[i];
    float4 v = ((const float4*)x)[(long long)i * 32 + c4];
    float* p = pooled + (long long)g * DIM + c4 * 4;
    atomicAdd(p + 0, v.x);
    atomicAdd(p + 1, v.y);
    atomicAdd(p + 2, v.z);
    atomicAdd(p + 3, v.w);
}

// ---------------- MLP head ----------------
__global__ void k_head1(const float* __restrict__ pooled, const float* __restrict__ cnt,
                        const float* __restrict__ W1, const float* __restrict__ b1,
                        float* __restrict__ h) {
    int idx = blockIdx.x * blockDim.x + threadIdx.x;   // 4096 * 64
    if (idx >= N_GRAPHS * HDIM) return;
    int g = idx >> 6;
    int j = idx & 63;
    float inv = 1.0f / fmaxf(cnt[g], 1.0f);
    const float* pg = pooled + (long long)g * DIM;
    float s = 0.f;
    #pragma unroll 4
    for (int c = 0; c < DIM; ++c) s += pg[c] * W1[c * HDIM + j];
    h[idx] = fmaxf(s * inv + b1[j], 0.f);
}

__global__ void k_head2(const float* __restrict__ h, const float* __restrict__ W2,
                        const float* __restrict__ b2, float* __restrict__ out) {
    int g = blockIdx.x * blockDim.x + threadIdx.x;
    if (g >= N_GRAPHS) return;
    const float* hg = h + (long long)g * HDIM;
    float s = 0.f;
    #pragma unroll 4
    for (int j = 0; j < HDIM; ++j) s += hg[j] * W2[j];
    out[g] = s + b2[0];
}

// ---------------- host orchestration ----------------
static inline int blks(long long n, int b) { return (int)((n + b - 1) / b); }

extern "C" void kernel_launch(void* const* d_in, const int* in_sizes, int n_in,
                              void* d_out, int out_size, void* d_ws, size_t ws_size,
                              hipStream_t stream) {
    const float* x_in   = (const float*)d_in[0];
    const float* Wl     = (const float*)d_in[1];
    const float* Wr     = (const float*)d_in[2];
    const float* b      = (const float*)d_in[3];
    const float* gamma  = (const float*)d_in[4];
    const float* beta   = (const float*)d_in[5];
    const float* hW1    = (const float*)d_in[6];
    const float* hb1    = (const float*)d_in[7];
    const float* hW2    = (const float*)d_in[8];
    const float* hb2    = (const float*)d_in[9];
    const int*   esrc   = (const int*)d_in[10];
    const int*   edst   = (const int*)d_in[11];
    const int*   batch  = (const int*)d_in[12];
    float* out = (float*)d_out;

    // workspace layout (float offsets)
    float* ws = (float*)d_ws;
    const long long NF = (long long)N_NODES * DIM;        // 12,800,000
    float* B0      = ws;                                   // x (current)
    float* B1      = ws + NF;                              // agg
    float* B2      = ws + 2 * NF;                          // y (pre-BN)
    float* invdeg  = ws + 3 * NF;                          // 100,000
    float* bnsum   = invdeg + N_NODES;                     // 256
    float* bnss    = bnsum + 2 * DIM;                      // 256
    float* pooled  = bnss + 2 * DIM;                       // 4096*128
    float* cnt     = pooled + (long long)N_GRAPHS * DIM;   // 4096
    float* hbuf    = cnt + N_GRAPHS;                       // 4096*64
    float* Wpack   = hbuf + (long long)N_GRAPHS * HDIM;    // 256*128 packed weights

    const long long n4 = (long long)N_NODES * 32;          // float4 count per node buffer

    // x -> B0 (never mutate inputs)
    k_copy4<<<blks(n4, 256), 256, 0, stream>>>((const float4*)x_in, (float4*)B0, n4);

    // in-degree -> inverse degree (constant across layers)
    k_zero<<<blks(N_NODES, 256), 256, 0, stream>>>(invdeg, N_NODES);
    k_deg<<<blks(N_EDGES, 256), 256, 0, stream>>>(edst, invdeg);
    k_invdeg<<<blks(N_NODES, 256), 256, 0, stream>>>(invdeg);

    const long long scatterThreads = (long long)N_EDGES * 32;
    const int gemmBlocks = blks(N_NODES, ROWS_PER_BLK);    // 1563

    for (int l = 0; l < N_LAYERS; ++l) {
        // mean aggregation
        k_zero<<<blks(NF, 256), 256, 0, stream>>>(B1, NF);
        k_scatter<<<blks(scatterThreads, 256), 256, 0, stream>>>(esrc, edst, B0, B1);

        // pack this layer's [Wl;Wr] into B-fragment order (L2-hot, 128 KB)
        k_packW<<<blks(64 * 8 * 32, 256), 256, 0, stream>>>(
            Wl + (long long)l * DIM * DIM, Wr + (long long)l * DIM * DIM,
            (float2*)Wpack);

        // fused fp32 WMMA GEMM: B2 = (B1*invdeg)@Wl[l] + B0@Wr[l] + b[l]
        k_gemm_wmma<<<gemmBlocks, 256, 0, stream>>>(
            B1, B0, invdeg, (const float2*)Wpack,
            b + (long long)l * DIM, B2);

        // BatchNorm (batch stats) + ReLU -> B0
        k_zero<<<1, 256, 0, stream>>>(bnsum, 2 * DIM);
        k_bnstats<<<512, DIM, 0, stream>>>(B2, bnsum);
        k_bnfinal<<<1, DIM, 0, stream>>>(bnsum, gamma + (long long)l * DIM,
                                         beta + (long long)l * DIM, bnss);
        k_bnapply<<<blks(n4, 256), 256, 0, stream>>>(B2, bnss, B0);
    }

    // global mean pool
    k_zero<<<blks((long long)N_GRAPHS * DIM + N_GRAPHS, 256), 256, 0, stream>>>(
        pooled, (long long)N_GRAPHS * DIM + N_GRAPHS);   // pooled + cnt contiguous
    k_cnt<<<blks(N_NODES, 256), 256, 0, stream>>>(batch, cnt);
    k_pool<<<blks(n4, 256), 256, 0, stream>>>(batch, B0, pooled);

    // head
    k_head1<<<blks((long long)N_GRAPHS * HDIM, 256), 256, 0, stream>>>(
        pooled, cnt, hW1, hb1, hbuf);
    k_head2<<<blks(N_GRAPHS, 256), 256, 0, stream>>>(hbuf, hW2, hb2, out);

    (void)in_sizes; (void)n_in; (void)out_size; (void)ws_size;
}